// AGNN_19782619365935
// MI455X (gfx1250) — compile-verified
//
#include <hip/hip_runtime.h>
#include <hip/hip_bf16.h>
#include <cstddef>

typedef __attribute__((ext_vector_type(2))) float v2f;
typedef __attribute__((ext_vector_type(8))) float v8f;

#define FEAT 128
#define HID 16
#define NCLS 20
#define EPSN 1e-12f

// ---------------------------------------------------------------------------
// K1: h = relu(emb[x_ids] @ W1 + b1)   via V_WMMA_F32_16X16X4_F32
// One wave (32 lanes) computes one 16-node tile; K=128 -> 32 chained WMMAs.
// ---------------------------------------------------------------------------
__global__ __launch_bounds__(256) void k_embed_gemm_relu(
    const int* __restrict__ x_ids, const float* __restrict__ emb,
    const float* __restrict__ W1, const float* __restrict__ b1,
    float* __restrict__ h, int n_nodes)
{
    const int lane = threadIdx.x & 31;
    const int wave = blockIdx.x * (blockDim.x >> 5) + (threadIdx.x >> 5);
    const int row0 = wave * 16;
    if (row0 >= n_nodes) return;           // whole-wave uniform: EXEC all-1s below

    const int m  = lane & 15;              // row (A) / col (B,D) index
    const int kb = (lane >> 4) << 1;       // K-pair base: 0 or 2

    int node = row0 + m;
    if (node >= n_nodes) node = n_nodes - 1;     // safe clamp (N%16==0 normally)
    const size_t id = (size_t)x_ids[node];
    const float* __restrict__ arow = emb + id * FEAT + kb;
    const float* __restrict__ brow = W1 + (size_t)kb * HID + m;  // stride 64 per K-step

    v8f acc = {};
#pragma unroll
    for (int s = 0; s < FEAT / 4; ++s) {
        v2f a, b;
        a.x = arow[4 * s];         // A[m][4s+kb]
        a.y = arow[4 * s + 1];     // A[m][4s+kb+1]
        b.x = brow[64 * s];        // W1[4s+kb][m]
        b.y = brow[64 * s + 16];   // W1[4s+kb+1][m]
        acc = __builtin_amdgcn_wmma_f32_16x16x4_f32(
            /*neg_a=*/false, a, /*neg_b=*/false, b,
            /*c_mod=*/(short)0, acc, /*reuse_a=*/false, /*reuse_b=*/false);
    }

    // D layout: VGPR r -> row r (lanes 0-15) / row r+8 (lanes 16-31), col = m
    const int rbase = row0 + ((lane >> 4) ? 8 : 0);
    const float bias = b1[m];
#pragma unroll
    for (int r = 0; r < 8; ++r) {
        const int rr = rbase + r;
        if (rr < n_nodes) {
            float v = acc[r] + bias;
            h[(size_t)rr * HID + m] = v > 0.f ? v : 0.f;
        }
    }
}

// ---------------------------------------------------------------------------
// K2: hn = h / max(||h||, eps)          one thread per node
// ---------------------------------------------------------------------------
__global__ __launch_bounds__(256) void k_normalize(
    const float* __restrict__ h, float* __restrict__ hn, int n)
{
    const int i = blockIdx.x * blockDim.x + threadIdx.x;
    if (i >= n) return;
    const float4* r = (const float4*)(h + (size_t)i * HID);
    float4 v0 = r[0], v1 = r[1], v2 = r[2], v3 = r[3];
    float ss = v0.x*v0.x + v0.y*v0.y + v0.z*v0.z + v0.w*v0.w
             + v1.x*v1.x + v1.y*v1.y + v1.z*v1.z + v1.w*v1.w
             + v2.x*v2.x + v2.y*v2.y + v2.z*v2.z + v2.w*v2.w
             + v3.x*v3.x + v3.y*v3.y + v3.z*v3.z + v3.w*v3.w;
    const float inv = 1.0f / fmaxf(sqrtf(ss), EPSN);
    float4* o = (float4*)(hn + (size_t)i * HID);
    v0.x*=inv; v0.y*=inv; v0.z*=inv; v0.w*=inv;
    v1.x*=inv; v1.y*=inv; v1.z*=inv; v1.w*=inv;
    v2.x*=inv; v2.y*=inv; v2.z*=inv; v2.w*=inv;
    v3.x*=inv; v3.y*=inv; v3.z*=inv; v3.w*=inv;
    o[0]=v0; o[1]=v1; o[2]=v2; o[3]=v3;
}

// ---------------------------------------------------------------------------
// K3: per edge  e = exp(beta * <hn[dst],hn[src]>);  s[dst] += e
// |alpha| <= |beta| ~ 1, so the segment-max shift is unnecessary (softmax is
// shift-invariant; exp(+-1) is numerically safe) -> one pass saved.
// Edges e >= E_real are the appended self-loops (src=dst=e-E_real).
// ---------------------------------------------------------------------------
__global__ __launch_bounds__(256) void k_edge_alpha(
    const int* __restrict__ src, const int* __restrict__ dst,
    const float* __restrict__ hn, float* __restrict__ e_ws,
    float* __restrict__ s, const float* __restrict__ beta_p,
    float beta_fixed, int use_ptr, int E_tot, int E_real)
{
    const int e = blockIdx.x * blockDim.x + threadIdx.x;
    if (e >= E_tot) return;
    int si, di;
    if (e < E_real) { si = src[e]; di = dst[e]; }
    else            { si = di = e - E_real; }
    const float beta = use_ptr ? beta_p[0] : beta_fixed;
    const float4* a = (const float4*)(hn + (size_t)si * HID);
    const float4* b = (const float4*)(hn + (size_t)di * HID);
    float d = 0.f;
#pragma unroll
    for (int k = 0; k < 4; ++k) {
        float4 av = a[k], bv = b[k];
        d += av.x*bv.x + av.y*bv.y + av.z*bv.z + av.w*bv.w;
    }
    const float ex = __expf(beta * d);
    e_ws[e] = ex;
    unsafeAtomicAdd(&s[di], ex);
}

// ---------------------------------------------------------------------------
// K4: per edge  out[dst] += (e / s[dst]) * x[src]      (16 fp32 atomics)
// ---------------------------------------------------------------------------
__global__ __launch_bounds__(256) void k_edge_aggr(
    const int* __restrict__ src, const int* __restrict__ dst,
    const float* __restrict__ x, const float* __restrict__ e_ws,
    const float* __restrict__ s, float* __restrict__ out,
    int E_tot, int E_real)
{
    const int e = blockIdx.x * blockDim.x + threadIdx.x;
    if (e >= E_tot) return;
    int si, di;
    if (e < E_real) { si = src[e]; di = dst[e]; }
    else            { si = di = e - E_real; }
    const float w = e_ws[e] / s[di];
    const float4* xr = (const float4*)(x + (size_t)si * HID);
    float* o = out + (size_t)di * HID;
#pragma unroll
    for (int k = 0; k < 4; ++k) {
        float4 v = xr[k];
        unsafeAtomicAdd(&o[4*k + 0], w * v.x);
        unsafeAtomicAdd(&o[4*k + 1], w * v.y);
        unsafeAtomicAdd(&o[4*k + 2], w * v.z);
        unsafeAtomicAdd(&o[4*k + 3], w * v.w);
    }
}

// ---------------------------------------------------------------------------
// K5: logits = x @ W2 + b2 ; out = log_softmax(logits)   one thread per node
// ---------------------------------------------------------------------------
__global__ __launch_bounds__(256) void k_head(
    const float* __restrict__ x, const float* __restrict__ W2,
    const float* __restrict__ b2, float* __restrict__ out, int n)
{
    const int i = blockIdx.x * blockDim.x + threadIdx.x;
    if (i >= n) return;
    float xi[HID];
    const float4* xr = (const float4*)(x + (size_t)i * HID);
#pragma unroll
    for (int k = 0; k < 4; ++k) {
        float4 v = xr[k];
        xi[4*k+0]=v.x; xi[4*k+1]=v.y; xi[4*k+2]=v.z; xi[4*k+3]=v.w;
    }
    float logit[NCLS];
    float mx = -1e30f;
#pragma unroll
    for (int c = 0; c < NCLS; ++c) {
        float acc = b2[c];
#pragma unroll
        for (int k = 0; k < HID; ++k) acc += xi[k] * W2[k * NCLS + c];
        logit[c] = acc;
        mx = fmaxf(mx, acc);
    }
    float ssum = 0.f;
#pragma unroll
    for (int c = 0; c < NCLS; ++c) ssum += __expf(logit[c] - mx);
    const float lse = mx + __logf(ssum);
    float* o = out + (size_t)i * NCLS;
#pragma unroll
    for (int c = 0; c < NCLS; ++c) o[c] = logit[c] - lse;
}

// ---------------------------------------------------------------------------
extern "C" void kernel_launch(void* const* d_in, const int* in_sizes, int n_in,
                              void* d_out, int out_size, void* d_ws, size_t ws_size,
                              hipStream_t stream)
{
    const int*   x_ids = (const int*)d_in[0];
    const int*   eidx  = (const int*)d_in[1];
    const float* emb   = (const float*)d_in[2];
    const float* W1    = (const float*)d_in[3];
    const float* b1    = (const float*)d_in[4];
    const float* beta2 = (const float*)d_in[5];
    const float* W2    = (const float*)d_in[6];
    const float* b2    = (const float*)d_in[7];
    float* outp = (float*)d_out;

    const int N    = in_sizes[0];          // x_ids has N*1 elements
    const int E    = in_sizes[1] / 2;      // edge_index is [2, E]
    const int Etot = E + N;                // + self loops
    const int* src = eidx;
    const int* dst = eidx + E;

    // workspace carve-up (256B aligned)
    char* ws = (char*)d_ws;
    size_t off = 0;
    auto carve = [&](size_t bytes) -> char* {
        char* p = ws + off;
        off += (bytes + 255) & ~(size_t)255;
        return p;
    };
    const int tiles = (N + 15) / 16;
    float* h    = (float*)carve((size_t)tiles * 16 * HID * sizeof(float));
    float* hn   = (float*)carve((size_t)N * HID * sizeof(float));
    float* out1 = (float*)carve((size_t)N * HID * sizeof(float));
    float* e_ws = (float*)carve((size_t)Etot * sizeof(float));
    float* s    = (float*)carve((size_t)N * sizeof(float));
    float* out2 = h;   // h is dead after prop1's aggregate; reuse for prop2 output

    const int nb = (N + 255) / 256;
    const int eb = (Etot + 255) / 256;
    const int gb = (tiles + 7) / 8;        // 8 waves (tiles) per 256-thread block

    // ---- embed + W1 GEMM (WMMA) + relu ----
    k_embed_gemm_relu<<<gb, 256, 0, stream>>>(x_ids, emb, W1, b1, h, N);

    // ---- prop 1 (beta = 1.0 fixed) ----
    hipMemsetAsync(s, 0, (size_t)N * sizeof(float), stream);
    hipMemsetAsync(out1, 0, (size_t)N * HID * sizeof(float), stream);
    k_normalize<<<nb, 256, 0, stream>>>(h, hn, N);
    k_edge_alpha<<<eb, 256, 0, stream>>>(src, dst, hn, e_ws, s, beta2, 1.0f, 0, Etot, E);
    k_edge_aggr <<<eb, 256, 0, stream>>>(src, dst, h, e_ws, s, out1, Etot, E);

    // ---- prop 2 (beta = beta2, from device scalar) ----
    hipMemsetAsync(s, 0, (size_t)N * sizeof(float), stream);
    hipMemsetAsync(out2, 0, (size_t)N * HID * sizeof(float), stream); // after h last read
    k_normalize<<<nb, 256, 0, stream>>>(out1, hn, N);
    k_edge_alpha<<<eb, 256, 0, stream>>>(src, dst, hn, e_ws, s, beta2, 1.0f, 1, Etot, E);
    k_edge_aggr <<<eb, 256, 0, stream>>>(src, dst, out1, e_ws, s, out2, Etot, E);

    // ---- head: W2/b2 + log_softmax ----
    k_head<<<nb, 256, 0, stream>>>(out2, W2, b2, outp, N);
}